// MarginRankingLoss_19138374271444
// MI455X (gfx1250) — compile-verified
//
#include <hip/hip_runtime.h>
#include <hip/hip_bf16.h>
#include <stdint.h>

// loss[m,n] = max(0, BIAS - 0.5*(w_m + w_n)*|p_m - p_n|)   (gt input provably unused:
// r only appears multiplied by diff, and the gt-dependent branch is taken iff diff==0)
// out = sum_{m,n} loss / (n*n)

#define BIAS  0.1f
#define BLOCK 256
#define CHUNK 512     // n-elements staged into LDS per block
#define MROWS 2       // m rows per thread

typedef __attribute__((ext_vector_type(2))) float v2f;
typedef __attribute__((ext_vector_type(8))) float v8f;

// ---- wave32 sum via V_WMMA_F32_16X16X4_F32 --------------------------------
// A (16x4 f32): lanes0-15 hold {K0,K1}, lanes16-31 hold {K2,K3}. Partial goes
// in the first component (K0/K2 cols), zeros elsewhere. B = ones(4x16) =>
// D[i,j] = A[i,0] + A[i,2] = part[i] + part[i+16], replicated over j.
// D layout: vgpr r, lanes0-15 -> row r; lanes16-31 -> row r+8. Sum of the 8 D
// regs gives rows0-7 / rows8-15 per lane half; one xor-16 shuffle combines.
__device__ __forceinline__ float wave32_sum(float v) {
#if __has_builtin(__builtin_amdgcn_wmma_f32_16x16x4_f32)
  v2f a; a[0] = v;    a[1] = 0.0f;
  v2f b; b[0] = 1.0f; b[1] = 1.0f;
  v8f c = {};
  v8f d = __builtin_amdgcn_wmma_f32_16x16x4_f32(
      /*neg_a=*/false, a, /*neg_b=*/false, b,
      /*c_mod=*/(short)0, c, /*reuse_a=*/false, /*reuse_b=*/false);
  float s = ((d[0] + d[1]) + (d[2] + d[3])) + ((d[4] + d[5]) + (d[6] + d[7]));
  s += __shfl_xor(s, 16);
  return s;
#else
  for (int off = 16; off > 0; off >>= 1) v += __shfl_xor(v, off);
  return v;
#endif
}

// Block-wide (256-thread) sum; valid on thread 0 only.
__device__ __forceinline__ float block_sum_256(float v, float* swave) {
  float s = wave32_sum(v);
  const int lane = threadIdx.x & 31;
  const int wid  = threadIdx.x >> 5;
  if (lane == 0) swave[wid] = s;
  __syncthreads();
  float tot = 0.0f;
  if (threadIdx.x == 0) {
#pragma unroll
    for (int i = 0; i < BLOCK / 32; ++i) tot += swave[i];
  }
  return tot;
}

__device__ __forceinline__ float pair_loss(float pn, float wn, float pm, float wm) {
  float dp  = pm - pn;
  float had = 0.5f * fabsf(dp);          // |src| is a free modifier
  float sw  = wm + wn;
  return fmaxf(fmaf(sw, -had, BIAS), 0.0f);
}

// ---- main pairwise kernel --------------------------------------------------
__global__ void __launch_bounds__(BLOCK)
pair_hinge_kernel(const float* __restrict__ pred,
                  const float* __restrict__ wgt,
                  float* __restrict__ partials, int n) {
  __shared__ __align__(16) float2 tile[CHUNK];  // {p_n, w_n} interleaved
  __shared__ float swave[BLOCK / 32];

  const int base = blockIdx.y * CHUNK;
  const int cnt  = min(CHUNK, n - base);

  // Stage this block's n-chunk into LDS with CDNA5 async global->LDS copies.
  for (int k = threadIdx.x; k < cnt; k += BLOCK) {
    unsigned lp = (unsigned)(uintptr_t)&tile[k].x;  // low 32b of generic = LDS offset
    unsigned lw = (unsigned)(uintptr_t)&tile[k].y;
    uint64_t gp = (uint64_t)(uintptr_t)(pred + base + k);
    uint64_t gw = (uint64_t)(uintptr_t)(wgt  + base + k);
    asm volatile("global_load_async_to_lds_b32 %0, %1, off"
                 :: "v"(lp), "v"(gp) : "memory");
    asm volatile("global_load_async_to_lds_b32 %0, %1, off"
                 :: "v"(lw), "v"(gw) : "memory");
  }
  asm volatile("s_wait_asynccnt 0" ::: "memory");
  __syncthreads();

  // Two m rows per thread: halves LDS traffic per pair, two independent FMA
  // chains. All lanes of a wave read the same tile address -> LDS broadcast.
  const int m0 = blockIdx.x * (MROWS * BLOCK) + threadIdx.x;
  const int m1 = m0 + BLOCK;
  const int c0 = min(m0, n - 1);     // clamp: no OOB reads; junk discarded below
  const int c1 = min(m1, n - 1);
  const float pm0 = pred[c0], wm0 = wgt[c0];
  const float pm1 = pred[c1], wm1 = wgt[c1];

  float acc0 = 0.0f, acc1 = 0.0f;
  const float4* t4 = (const float4*)tile;      // 1x ds_load_b128 = 2 pairs
  const int nq = cnt >> 2;                     // iterations of 4 pairs
  for (int q = 0; q < nq; ++q) {
    float4 A = t4[2 * q];
    float4 B = t4[2 * q + 1];
    acc0 += pair_loss(A.x, A.y, pm0, wm0);  acc1 += pair_loss(A.x, A.y, pm1, wm1);
    acc0 += pair_loss(A.z, A.w, pm0, wm0);  acc1 += pair_loss(A.z, A.w, pm1, wm1);
    acc0 += pair_loss(B.x, B.y, pm0, wm0);  acc1 += pair_loss(B.x, B.y, pm1, wm1);
    acc0 += pair_loss(B.z, B.w, pm0, wm0);  acc1 += pair_loss(B.z, B.w, pm1, wm1);
  }
  for (int j = nq << 2; j < cnt; ++j) {        // remainder (empty when cnt%4==0)
    float2 t = tile[j];
    acc0 += pair_loss(t.x, t.y, pm0, wm0);
    acc1 += pair_loss(t.x, t.y, pm1, wm1);
  }
  if (m0 >= n) acc0 = 0.0f;
  if (m1 >= n) acc1 = 0.0f;

  float tot = block_sum_256(acc0 + acc1, swave);
  if (threadIdx.x == 0)
    partials[blockIdx.y * gridDim.x + blockIdx.x] = tot;
}

// ---- deterministic finalize ------------------------------------------------
__global__ void __launch_bounds__(BLOCK)
finalize_kernel(const float* __restrict__ partials, float* __restrict__ out,
                int nparts, int n) {
  __shared__ float swave[BLOCK / 32];
  float s = 0.0f;
  for (int i = threadIdx.x; i < nparts; i += BLOCK) s += partials[i];
  float tot = block_sum_256(s, swave);
  if (threadIdx.x == 0) {
    float denom = (float)n * (float)n;   // 2^26 for B=8192, exact
    out[0] = tot / denom;
  }
}

extern "C" void kernel_launch(void* const* d_in, const int* in_sizes, int n_in,
                              void* d_out, int out_size, void* d_ws, size_t ws_size,
                              hipStream_t stream) {
  const float* pred = (const float*)d_in[0];
  // d_in[1] (correct_output) provably does not affect the result.
  const float* wgt  = (const float*)d_in[2];
  const int n = in_sizes[0];

  const int mtiles  = (n + MROWS * BLOCK - 1) / (MROWS * BLOCK);  // 16 for B=8192
  const int nchunks = (n + CHUNK - 1) / CHUNK;                    // 16 for B=8192
  float* partials = (float*)d_ws;            // mtiles*nchunks floats (1 KB)

  dim3 grid(mtiles, nchunks);
  pair_hinge_kernel<<<grid, BLOCK, 0, stream>>>(pred, wgt, partials, n);
  finalize_kernel<<<1, BLOCK, 0, stream>>>(partials, (float*)d_out,
                                           mtiles * nchunks, n);
}